// HybridLoss_10677288698224
// MI455X (gfx1250) — compile-verified
//
#include <hip/hip_runtime.h>

typedef __attribute__((ext_vector_type(2))) float v2f;
typedef __attribute__((ext_vector_type(8))) float v8f;

#define NSZ 4096
#define NSLICE 8   // B*T = 2*4

#define BIGF    1.0e10f
#define THRESHF 1.0e9f
#define NEGF   -1.0e10f
#define ALPHAF  0.5f
#define TEMPF   0.1f

// ---------------------------------------------------------------------------
// Kernel 1: reset workspace (min arrays to bits(BIG)) and zero the output.
// Runs every launch so graph replay is deterministic.
// ---------------------------------------------------------------------------
__global__ void hl_init_kernel(unsigned int* __restrict__ minp,
                               unsigned int* __restrict__ mint,
                               float* __restrict__ out) {
    int gid = blockIdx.x * blockDim.x + threadIdx.x;
    unsigned int big = __float_as_uint(BIGF);
    if (gid < NSLICE * NSZ) {
        minp[gid] = big;
        mint[gid] = big;
    }
    if (gid == 0) out[0] = 0.0f;
}

// ---------------------------------------------------------------------------
// Kernel 2: tiled pairwise distance + masked row/col min.
// Masks are pre-baked into xx/yy (unmasked -> BIG, which dominates exactly in
// f32 since ulp(1e10)=1024 >> |yy - 2xy|), so the hot loop is branch-free.
// Two WMMAs per 16x16 subtile (P*Q^T and Q*P^T) so BOTH min directions are
// per-lane register accumulations; A(16x4) and B(4x16) share the same
// per-lane layout, so one v2f per point serves both operand roles.
// Block = 256 threads (8 waves), tile = 128x128; wave w owns 16-col strip w.
// ---------------------------------------------------------------------------
__global__ __launch_bounds__(256)
void hl_dist_kernel(const float* __restrict__ pred,
                    const float* __restrict__ tgt,
                    const unsigned char* __restrict__ ctrl,
                    const unsigned char* __restrict__ pad,
                    const unsigned char* __restrict__ vis,
                    unsigned int* __restrict__ minp,
                    unsigned int* __restrict__ mint) {
    const int slice = blockIdx.z;          // b*T + t
    const int b     = slice >> 2;          // T == 4
    const int m0    = blockIdx.y * 128;    // pred row tile base
    const int n0    = blockIdx.x * 128;    // target col tile base

    const float* P = pred + (size_t)slice * NSZ * 3;
    const float* Q = tgt  + (size_t)slice * NSZ * 3;

    __shared__ float        xx_s[128];     // mask ? |p|^2 : BIG
    __shared__ float        yy_s[128];     // mask ? |q|^2 : BIG
    __shared__ unsigned int tminp[128];
    __shared__ unsigned int tmint[128];

    const int tid = threadIdx.x;
    const unsigned int bigb = __float_as_uint(BIGF);

    if (tid < 128) {
        int n = m0 + tid;
        float x = P[n * 3 + 0], y = P[n * 3 + 1], z = P[n * 3 + 2];
        bool m = (ctrl[b * NSZ + n] == 0) && pad[b * NSZ + n] && vis[slice * NSZ + n];
        xx_s[tid]  = m ? (x * x + y * y + z * z) : BIGF;
        tminp[tid] = bigb;
    } else {
        int c = tid - 128;
        int n = n0 + c;
        float x = Q[n * 3 + 0], y = Q[n * 3 + 1], z = Q[n * 3 + 2];
        bool m = (ctrl[b * NSZ + n] == 0) && pad[b * NSZ + n] && vis[slice * NSZ + n];
        yy_s[c]  = m ? (x * x + y * y + z * z) : BIGF;
        tmint[c] = bigb;
    }
    __syncthreads();

    const int wave = tid >> 5;     // 0..7 : subtile column strip
    const int lane = tid & 31;
    const int hi   = lane >> 4;    // lane half selects K pair (K=0,1 vs K=2,3)
    const int l    = lane & 15;

    // Target point for this wave's column l (serves as B in acc1 and A in acc2).
    // Layout: lanes 0-15 carry K=0,1 = (x,y); lanes 16-31 carry K=2,3 = (z,0).
    const int cG = n0 + wave * 16 + l;
    v2f qv;
    qv.x = Q[cG * 3 + 2 * hi];
    { float t = Q[cG * 3 + 1 + hi]; qv.y = hi ? 0.0f : t; }

    const float  yyl = yy_s[wave * 16 + l];          // acc1 epilogue (fixed col)
    const float4* xx4 = (const float4*)xx_s;
    const float4* yy4 = (const float4*)yy_s;
    const float4 yw0 = yy4[wave * 4 + hi * 2 + 0];   // yy for tgt idx wave*16+hi*8+0..7
    const float4 yw1 = yy4[wave * 4 + hi * 2 + 1];
    const float yy2v[8] = { yw0.x, yw0.y, yw0.z, yw0.w, yw1.x, yw1.y, yw1.z, yw1.w };

    float cmin = BIGF;   // running min_t for this lane's target column

#pragma unroll
    for (int sr = 0; sr < 8; ++sr) {
        // Pred point for row l of subtile sr (A in acc1, B in acc2).
        const int rG = m0 + sr * 16 + l;
        v2f pv;
        pv.x = P[rG * 3 + 2 * hi];
        { float t = P[rG * 3 + 1 + hi]; pv.y = hi ? 0.0f : t; }

        v8f a1 = {};   // (pred rows) x (tgt cols): lane dim = tgt col
        a1 = __builtin_amdgcn_wmma_f32_16x16x4_f32(
            false, pv, false, qv, (short)0, a1, false, false);
        v8f a2 = {};   // (tgt rows) x (pred cols): lane dim = pred row
        a2 = __builtin_amdgcn_wmma_f32_16x16x4_f32(
            false, qv, false, pv, (short)0, a2, false, false);

        const float4 xw0 = xx4[sr * 4 + hi * 2 + 0];  // xx for rows sr*16+hi*8+0..7
        const float4 xw1 = xx4[sr * 4 + hi * 2 + 1];
        const float xx1v[8] = { xw0.x, xw0.y, xw0.z, xw0.w, xw1.x, xw1.y, xw1.z, xw1.w };
        const float xxl = xx_s[sr * 16 + l];          // acc2 epilogue (fixed row)

        float rmin = BIGF;
#pragma unroll
        for (int r = 0; r < 8; ++r) {
            // acc1: row = sr*16+hi*8+r, col = wave*16+l  -> feeds min_t (cmin)
            float d1 = fmaxf(xx1v[r] + yyl - 2.0f * a1[r], 0.0f);
            cmin = fminf(cmin, d1);
            // acc2: tgt = wave*16+hi*8+r, row = sr*16+l  -> feeds min_p (rmin)
            float d2 = fmaxf(yy2v[r] + xxl - 2.0f * a2[r], 0.0f);
            rmin = fminf(rmin, d2);
        }
        atomicMin(&tminp[sr * 16 + l], __float_as_uint(rmin));   // 2-way conflict only
    }
    atomicMin(&tmint[wave * 16 + l], __float_as_uint(cmin));     // 2-way conflict only
    __syncthreads();

    if (tid < 128) atomicMin(&minp[slice * NSZ + m0 + tid], tminp[tid]);
    else           atomicMin(&mint[slice * NSZ + n0 + (tid - 128)], tmint[tid - 128]);
}

// ---------------------------------------------------------------------------
// Kernel 3: per-slice Chamfer + soft-Hausdorff reduction, one block per slice.
// ---------------------------------------------------------------------------
__global__ __launch_bounds__(256)
void hl_finalize_kernel(const unsigned char* __restrict__ ctrl,
                        const unsigned char* __restrict__ pad,
                        const unsigned char* __restrict__ vis,
                        const unsigned int* __restrict__ minp,
                        const unsigned int* __restrict__ mint,
                        float* __restrict__ out) {
    const int slice = blockIdx.x;
    const int b     = slice >> 2;
    const int tid   = threadIdx.x;

    __shared__ float red[256];

    auto redsum = [&](float v) -> float {
        red[tid] = v; __syncthreads();
        for (int s = 128; s > 0; s >>= 1) {
            if (tid < s) red[tid] += red[tid + s];
            __syncthreads();
        }
        float r = red[0]; __syncthreads(); return r;
    };
    auto redmax = [&](float v) -> float {
        red[tid] = v; __syncthreads();
        for (int s = 128; s > 0; s >>= 1) {
            if (tid < s) red[tid] = fmaxf(red[tid], red[tid + s]);
            __syncthreads();
        }
        float r = red[0]; __syncthreads(); return r;
    };

    // Pass 1: mask count, chamfer sums, max of hp/ht for stable softmax.
    float cnt = 0.0f, cdp = 0.0f, cdt = 0.0f, mhp = NEGF, mht = NEGF;
    for (int n = tid; n < NSZ; n += 256) {
        const bool m = (ctrl[b * NSZ + n] == 0) && pad[b * NSZ + n] && vis[slice * NSZ + n];
        const float mp = __uint_as_float(minp[slice * NSZ + n]);
        const float mt = __uint_as_float(mint[slice * NSZ + n]);
        const float mf = m ? 1.0f : 0.0f;
        cnt += mf;
        cdp += (mp > THRESHF ? 0.0f : mp) * mf;
        cdt += (mt > THRESHF ? 0.0f : mt) * mf;
        mhp = fmaxf(mhp, m ? fminf(mp, THRESHF) : NEGF);
        mht = fmaxf(mht, m ? fminf(mt, THRESHF) : NEGF);
    }
    const float tcnt = redsum(cnt);
    const float tcdp = redsum(cdp);
    const float tcdt = redsum(cdt);
    const float gmhp = redmax(mhp);
    const float gmht = redmax(mht);

    // Pass 2: softmax-weighted sums (max-subtracted, matching jax.nn.softmax).
    float sp = 0.0f, sxp = 0.0f, st = 0.0f, sxt = 0.0f;
    const float invT = 1.0f / TEMPF;
    for (int n = tid; n < NSZ; n += 256) {
        const bool m = (ctrl[b * NSZ + n] == 0) && pad[b * NSZ + n] && vis[slice * NSZ + n];
        const float mp = __uint_as_float(minp[slice * NSZ + n]);
        const float mt = __uint_as_float(mint[slice * NSZ + n]);
        const float hp = m ? fminf(mp, THRESHF) : NEGF;
        const float ht = m ? fminf(mt, THRESHF) : NEGF;
        const float wp = __expf((hp - gmhp) * invT);
        const float wt = __expf((ht - gmht) * invT);
        sp += wp; sxp += hp * wp;
        st += wt; sxt += ht * wt;
    }
    const float tsp  = redsum(sp);
    const float tsxp = redsum(sxp);
    const float tst  = redsum(st);
    const float tsxt = redsum(sxt);

    if (tid == 0) {
        const bool hasv = tcnt > 0.0f;   // has_valid == any(mask)
        const float rp = hasv ? (tsxp / tsp) : 0.0f;
        const float rt = hasv ? (tsxt / tst) : 0.0f;
        const float hd = fmaxf(rp, rt);
        const float nv = fmaxf(tcnt, 1.0f);
        const float cd = (tcdp + tcdt) / nv;
        const float v  = ALPHAF * cd + (1.0f - ALPHAF) * hd;
        atomicAdd(out, v * (1.0f / (float)NSLICE));   // mean over B*T slices
    }
}

// ---------------------------------------------------------------------------
extern "C" void kernel_launch(void* const* d_in, const int* in_sizes, int n_in,
                              void* d_out, int out_size, void* d_ws, size_t ws_size,
                              hipStream_t stream) {
    const float* pred         = (const float*)d_in[0];         // (2,4,4096,3) f32
    const float* tgt          = (const float*)d_in[1];         // (2,4,4096,3) f32
    const unsigned char* ctrl = (const unsigned char*)d_in[2]; // (2,4096) bool
    const unsigned char* pad  = (const unsigned char*)d_in[3]; // (2,4096) bool
    const unsigned char* vis  = (const unsigned char*)d_in[4]; // (2,4,4096) bool
    float* out = (float*)d_out;

    unsigned int* minp = (unsigned int*)d_ws;          // NSLICE*NSZ u32
    unsigned int* mint = minp + NSLICE * NSZ;          // NSLICE*NSZ u32

    // 1) init workspace + output
    hl_init_kernel<<<(NSLICE * NSZ + 255) / 256, 256, 0, stream>>>(minp, mint, out);

    // 2) tiled WMMA distance + masked min reduction
    dim3 grid(NSZ / 128, NSZ / 128, NSLICE);
    hl_dist_kernel<<<grid, 256, 0, stream>>>(pred, tgt, ctrl, pad, vis, minp, mint);

    // 3) per-slice chamfer + soft hausdorff, accumulate scalar
    hl_finalize_kernel<<<NSLICE, 256, 0, stream>>>(ctrl, pad, vis, minp, mint, out);
}